// HyperConnections_37056977830035
// MI455X (gfx1250) — compile-verified
//
#include <hip/hip_runtime.h>
#include <math.h>

// Problem constants (B,S,N,D) = (4,4,2048,2048), T = S+1 = 5 (pad to 6 with beta col)
#define NB 4
#define NS 4
#define NN 2048
#define ND 2048

typedef __attribute__((ext_vector_type(2))) float v2f;
typedef __attribute__((ext_vector_type(4))) float v4f;
typedef __attribute__((ext_vector_type(8))) float v8f;

// ---------------------------------------------------------------------------
// Prep: build per-lane WMMA B-fragments for Bw[k][t] = (gamma[k]+1) * w_t[k]
//   t = 0..4 : w_alpha rows ; t = 5 : w_beta ; t = 6..15 : zero padding
// Fragment layout for V_WMMA_F32_16X16X4_F32 B (4x16, K x N):
//   lane l: N = l%16 ; VGPR0 = B[kA][N], VGPR1 = B[kA+1][N], kA = k0 + 2*(l/16)
// Stored flat: frag i (k0 = 4*i), lane l -> float2 at index (i*32 + l)
// ---------------------------------------------------------------------------
__global__ __launch_bounds__(256) void hc_prep_bw(const float* __restrict__ gamma,
                                                  const float* __restrict__ w_alpha,
                                                  const float* __restrict__ w_beta,
                                                  float* __restrict__ bw) {
    int gid = blockIdx.x * blockDim.x + threadIdx.x;  // 512*32 = 16384 fragment-lanes
    if (gid >= (ND / 4) * 32) return;
    int i = gid >> 5;
    int l = gid & 31;
    int t = l & 15;
    int k = i * 4 + ((l >> 4) << 1);
    float v0 = 0.f, v1 = 0.f;
    if (t < 5) {
        v0 = (gamma[k] + 1.f) * w_alpha[t * ND + k];
        v1 = (gamma[k + 1] + 1.f) * w_alpha[t * ND + k + 1];
    } else if (t == 5) {
        v0 = (gamma[k] + 1.f) * w_beta[k];
        v1 = (gamma[k + 1] + 1.f) * w_beta[k + 1];
    }
    bw[2 * gid]     = v0;
    bw[2 * gid + 1] = v1;
}

// ---------------------------------------------------------------------------
// Main: one workgroup = (b, 16 consecutive n rows). 256 threads = 8 waves:
//   wave = s (0..3) x K-half (0..1).
// Phase 1: WMMA f32 16x16x4 GEMM  C[16n x 16t] = x[16n x K] * Bw[K x 16t]
//          + VALU sum(x^2); results -> LDS.  (RT loads: tile must stay in L2.)
// Phase 2: per-n alpha(4x5)/beta(4) from LDS, coalesced float4 reload of x
//          (non-temporal: last use), compute mix + beta-fused outputs
//          (non-temporal stores: write-once stream, keep L2 for residuals).
// ---------------------------------------------------------------------------
__global__ __launch_bounds__(256) void hc_main(const float* __restrict__ res,
                                               const float* __restrict__ bw,
                                               const float* __restrict__ scale_alpha,
                                               const float* __restrict__ static_alpha,
                                               const float* __restrict__ scale_beta,
                                               const float* __restrict__ static_beta,
                                               float* __restrict__ out) {
    __shared__ float rdot[2][NS][16][6];
    __shared__ float ssl[2][NS][16];
    __shared__ float ab[16][24];  // per n: [s*5+t] alpha (20), [20+s] beta (4)

    const int tid  = threadIdx.x;
    const int lane = tid & 31;
    const int wave = tid >> 5;
    const int s    = wave & 3;
    const int half = wave >> 2;
    const int b    = blockIdx.x >> 7;          // 128 n-tiles per b
    const int n0   = (blockIdx.x & 127) << 4;  // 16 n rows per tile

    const int nLoc = lane & 15;           // A-fragment M index (n within tile)
    const int kOff = (lane >> 4) << 1;    // lane-half K offset (0 or 2)

    // ---- Phase 1: K-loop of WMMA f32 16x16x4 -------------------------------
    v8f c0 = {};
    v8f c1 = {};
    float ss = 0.f;

    const size_t aBase =
        ((size_t)(b * NS + s) * NN + (size_t)(n0 + nLoc)) * ND + (size_t)(half * 1024) + kOff;
    const float* aPtr = res + aBase;
    const v2f*   bPtr = ((const v2f*)bw) + (size_t)half * 256 * 32 + lane;

    // 256 fragments of K=4 per wave (half of D=2048)
    #pragma unroll 4
    for (int i = 0; i < 256; i += 2) {
        v2f a0 = *(const v2f*)(aPtr + (size_t)i * 4);
        v2f a1 = *(const v2f*)(aPtr + (size_t)i * 4 + 4);
        v2f b0 = bPtr[(size_t)i * 32];
        v2f b1 = bPtr[(size_t)(i + 1) * 32];
        ss += a0.x * a0.x + a0.y * a0.y;
        ss += a1.x * a1.x + a1.y * a1.y;
        c0 = __builtin_amdgcn_wmma_f32_16x16x4_f32(false, a0, false, b0, (short)0, c0, false, false);
        c1 = __builtin_amdgcn_wmma_f32_16x16x4_f32(false, a1, false, b1, (short)0, c1, false, false);
    }

    // lane l and l^16 cover complementary (k mod 4) halves of the same n row
    ss += __shfl_xor(ss, 16);

    const int t = lane & 15;
    if (t < 6) {
        #pragma unroll
        for (int j = 0; j < 8; ++j) {
            int nn = j + ((lane >> 4) << 3);  // C layout: VGPR j -> M = j (+8 for hi lanes)
            rdot[half][s][nn][t] = c0[j] + c1[j];
        }
    }
    if (lane < 16) ssl[half][s][nLoc] = ss;
    __syncthreads();

    // ---- alpha/beta: 16n * 4s * 6t = 384 work items ------------------------
    const float sA = scale_alpha[0];
    const float sB = scale_beta[0];
    for (int w = tid; w < 16 * NS * 6; w += 256) {
        int n  = w / 24;
        int r  = w % 24;
        int s2 = r / 6;
        int tt = r % 6;
        float sq  = ssl[0][s2][n] + ssl[1][s2][n];
        float inv = 45.25483399593904f / fmaxf(sqrtf(sq), 1e-12f);  // sqrt(2048)/max(||x||,eps)
        float rv  = rdot[0][s2][n][tt] + rdot[1][s2][n][tt];
        float v   = tanhf(rv * inv);
        if (tt < 5)
            ab[n][s2 * 5 + tt] = v * sA + static_alpha[s2 * 5 + tt];
        else
            ab[n][20 + s2] = v * sB + static_beta[s2];
    }
    __syncthreads();

    // ---- Phase 2: mix + beta-fused output (x re-read hits L2) --------------
    const size_t strideS = (size_t)NN * ND;
    for (int n = 0; n < 16; ++n) {
        float Al[4][5];
        float Bt[4];
        #pragma unroll
        for (int s2 = 0; s2 < 4; ++s2) {
            #pragma unroll
            for (int tt = 0; tt < 5; ++tt) Al[s2][tt] = ab[n][s2 * 5 + tt];
            Bt[s2] = ab[n][20 + s2];
        }
        const size_t base0 = ((size_t)(b * NS) * NN + (size_t)(n0 + n)) * ND;
        #pragma unroll
        for (int it = 0; it < 2; ++it) {
            int d = it * 1024 + tid * 4;
            // last use of x: non-temporal loads so we don't evict other tiles
            v4f x0 = __builtin_nontemporal_load((const v4f*)(res + base0 + 0 * strideS + d));
            v4f x1 = __builtin_nontemporal_load((const v4f*)(res + base0 + 1 * strideS + d));
            v4f x2 = __builtin_nontemporal_load((const v4f*)(res + base0 + 2 * strideS + d));
            v4f x3 = __builtin_nontemporal_load((const v4f*)(res + base0 + 3 * strideS + d));
            v4f mix0 = Al[0][0] * x0 + Al[1][0] * x1 + Al[2][0] * x2 + Al[3][0] * x3;
            #pragma unroll
            for (int sp = 0; sp < 4; ++sp) {
                v4f o = Bt[sp] * mix0 + Al[0][sp + 1] * x0 + Al[1][sp + 1] * x1 +
                        Al[2][sp + 1] * x2 + Al[3][sp + 1] * x3;
                // write-once output stream: non-temporal store keeps L2 for residuals
                __builtin_nontemporal_store(o, (v4f*)(out + base0 + (size_t)sp * strideS + d));
            }
        }
    }
}

extern "C" void kernel_launch(void* const* d_in, const int* in_sizes, int n_in,
                              void* d_out, int out_size, void* d_ws, size_t ws_size,
                              hipStream_t stream) {
    (void)in_sizes; (void)n_in; (void)out_size; (void)ws_size;
    const float* residuals    = (const float*)d_in[0];
    const float* gamma        = (const float*)d_in[1];
    const float* w_alpha      = (const float*)d_in[2];
    const float* scale_alpha  = (const float*)d_in[3];
    const float* static_alpha = (const float*)d_in[4];
    const float* w_beta       = (const float*)d_in[5];
    const float* scale_beta   = (const float*)d_in[6];
    const float* static_beta  = (const float*)d_in[7];
    float* bw = (float*)d_ws;  // 512 fragments * 32 lanes * 2 floats = 128 KB

    hc_prep_bw<<<64, 256, 0, stream>>>(gamma, w_alpha, w_beta, bw);
    hc_main<<<NB * (NN / 16), 256, 0, stream>>>(residuals, bw, scale_alpha, static_alpha,
                                                scale_beta, static_beta, (float*)d_out);
}